// BennaSynapse_34737695490801
// MI455X (gfx1250) — compile-verified
//
#include <hip/hip_runtime.h>
#include <math.h>

#define TW1 4096
#define TW2 4096
#define NELEM ((size_t)TW1 * (size_t)TW2)

typedef float v2f __attribute__((ext_vector_type(2)));
typedef float v4f __attribute__((ext_vector_type(4)));
typedef float v8f __attribute__((ext_vector_type(8)));

#if defined(__HIP_DEVICE_COMPILE__) && __has_builtin(__builtin_amdgcn_wmma_f32_16x16x4_f32)
#define HAVE_WMMA_F32K4 1
#endif

// Branch-free tanh: 1 - 2/(exp(2x)+1) via hardware exp2/rcp (trans ops).
// Exact at +-saturation, ~1e-7 abs error elsewhere.
__device__ __forceinline__ float fast_tanh(float x) {
  const float e = __builtin_amdgcn_exp2f(x * 2.8853900817779268f); // 2*log2(e)
  const float r = __builtin_amdgcn_rcpf(e + 1.0f);
  return __builtin_fmaf(-2.0f, r, 1.0f);
}

// ---------------------------------------------------------------------------
// Pass 1: single sweep over W computing
//   r[k]  = sum_j a1[j] * W[j,k]
//   s67   = sum_{j,k} a1[j] * W[j,k] * e0[k]
//   s8    = sum_{j,k} e1[j] * W[j,k] * a0[k]
// Column-parallel, b128 loads (perfectly coalesced). Grid: 4 col-chunks x
// 32 row-chunks = 128 blocks of 256 threads; each thread owns 4 columns.
// ---------------------------------------------------------------------------
__global__ void reduce_pass(const float* __restrict__ Wm,
                            const float* __restrict__ a1,
                            const float* __restrict__ e1,
                            const float* __restrict__ a0,
                            const float* __restrict__ e0,
                            float* __restrict__ r,
                            float* __restrict__ scal /*[0]=s67 [1]=s8*/) {
  const int tid      = threadIdx.x;
  const int colchunk = blockIdx.x & 3;
  const int rowchunk = blockIdx.x >> 2;
  const int k        = colchunk * 1024 + tid * 4;
  const int j0       = rowchunk * 128;

  const v4f a0v = *(const v4f*)(a0 + k);
  const v4f e0v = *(const v4f*)(e0 + k);
  v4f accr = (v4f)(0.0f);
  float s67 = 0.f, s8 = 0.f;

#pragma unroll 2
  for (int j = j0; j < j0 + 128; ++j) {
    const v4f w  = *(const v4f*)(Wm + (size_t)j * TW2 + k);
    const float a1j = a1[j];
    const float e1j = e1[j];
    accr.x += a1j * w.x; accr.y += a1j * w.y;
    accr.z += a1j * w.z; accr.w += a1j * w.w;
    const float we0 = w.x * e0v.x + w.y * e0v.y + w.z * e0v.z + w.w * e0v.w;
    const float wa0 = w.x * a0v.x + w.y * a0v.y + w.z * a0v.z + w.w * a0v.w;
    s67 += a1j * we0;
    s8  += e1j * wa0;
  }

  atomicAdd(&r[k + 0], accr.x);
  atomicAdd(&r[k + 1], accr.y);
  atomicAdd(&r[k + 2], accr.z);
  atomicAdd(&r[k + 3], accr.w);

  // wave32 reduce, then cross-wave via LDS, one atomic per block per scalar
  for (int off = 16; off > 0; off >>= 1) {
    s67 += __shfl_down(s67, off);
    s8  += __shfl_down(s8, off);
  }
  __shared__ float red[16];
  const int lane = tid & 31, wv = tid >> 5;
  if (lane == 0) { red[wv] = s67; red[8 + wv] = s8; }
  __syncthreads();
  if (tid == 0) {
    float t = 0.f, u = 0.f;
    for (int i = 0; i < 8; ++i) { t += red[i]; u += red[8 + i]; }
    atomicAdd(&scal[0], t);
    atomicAdd(&scal[1], u);
  }
}

// ---------------------------------------------------------------------------
// Pass 2: compute s5 = sum(a1), fold scalars into per-column vectors
//   x[k] = cA*a0[k] + cC*e0[k]
//   y[k] = cB*e0[k] + P9*(a0[k] - r[k])
//   z[k] = cE*e0[k]
// ---------------------------------------------------------------------------
__global__ void prep_pass(const float* __restrict__ a0,
                          const float* __restrict__ e0,
                          const float* __restrict__ a1,
                          const float* __restrict__ P,
                          const float* __restrict__ r,
                          const float* __restrict__ scal,
                          float* __restrict__ xcol,
                          float* __restrict__ ycol,
                          float* __restrict__ zcol) {
  __shared__ float sh[256];
  const int tid = threadIdx.x;
  float s = 0.f;
  for (int j = tid; j < TW1; j += 256) s += a1[j];
  sh[tid] = s;
  __syncthreads();
  for (int off = 128; off > 0; off >>= 1) {
    if (tid < off) sh[tid] += sh[tid + off];
    __syncthreads();
  }
  const float s5  = sh[0];
  const float s67 = scal[0];
  const float s8  = scal[1];
  const float cA = -(P[0] + P[5] * s5  + P[7] * s67);
  const float cB = -(P[1] + P[6] * s67 + P[8] * s8);
  const float cC = -P[2];
  const float cE = -P[4];
  const float p9 =  P[9];
  for (int k = tid; k < TW2; k += 256) {
    const float a0k = a0[k], e0k = e0[k], rk = r[k];
    xcol[k] = cA * a0k + cC * e0k;
    ycol[k] = cB * e0k + p9 * (a0k - rk);
    zcol[k] = cE * e0k;
  }
}

// ---------------------------------------------------------------------------
// Pass 3: one wave per 16x16 tile.
//   Phase A: rank-3 bilinear term as one exact f32 V_WMMA_F32_16X16X4_F32:
//            U = A(16x4) x B(4x16),
//            A rows m: [e1[J0+m], a1[J0+m], 1, 0]
//            B rows:   [x[K0..]; z[K0..]; y[K0..]; 0]
//   Phase B: U tile bounced through LDS (stride-20 rows: conflict-free,
//            16B-aligned) so the global-memory phase runs entirely on
//            b128 loads / nontemporal b128 stores: 24 vmem ops per wave
//            instead of 96.
// ---------------------------------------------------------------------------
#define USTRIDE 20   // floats per tile row in LDS (16B-aligned, bank-clean)

__global__ void benna_main(const float* __restrict__ Wm,
                           const float* __restrict__ bias,
                           const float* __restrict__ chem,
                           const float* __restrict__ a1,
                           const float* __restrict__ e1,
                           const float* __restrict__ xcol,
                           const float* __restrict__ ycol,
                           const float* __restrict__ zcol,
                           const float* __restrict__ P,
                           const float* __restrict__ Cv,
                           const float* __restrict__ Gv,
                           float* __restrict__ out) {
  __shared__ float utile[8][16 * USTRIDE];   // 10 KB / block

  const int lane = threadIdx.x & 31;
  const int wave = threadIdx.x >> 5;
  const int tile = blockIdx.x * 8 + wave;
  const int J0   = (tile >> 8) << 4;   // 256 tiles per row band
  const int K0   = (tile & 255) << 4;
  const int n    = lane & 15;
  const bool hi  = lane >= 16;

  // Unconditional in-bounds loads, lane-select into WMMA operand layout.
  const float e1v = e1[J0 + n];
  const float a1v = a1[J0 + n];
  const float xv  = xcol[K0 + n];
  const float yv  = ycol[K0 + n];
  const float zv  = zcol[K0 + n];

  v8f U = {};
#ifdef HAVE_WMMA_F32K4
  v2f A, B;
  A.x = hi ? 1.0f : e1v;   // A[m][0]=e1 | A[m][2]=1
  A.y = hi ? 0.0f : a1v;   // A[m][1]=a1 | A[m][3]=0
  B.x = hi ? yv : xv;      // B[0][n]=x  | B[2][n]=y
  B.y = hi ? 0.0f : zv;    // B[1][n]=z  | B[3][n]=0
  U = __builtin_amdgcn_wmma_f32_16x16x4_f32(false, A, false, B,
                                            (short)0, U, false, false);
#else
#pragma unroll
  for (int v = 0; v < 8; ++v) {
    const int jm = J0 + v + (hi ? 8 : 0);
    U[v] = e1[jm] * xv + a1[jm] * yv + zv;
  }
#endif

  // D layout -> LDS tile: element (M = v + 8*hi, N = lane&15)
#pragma unroll
  for (int v = 0; v < 8; ++v) {
    const int m = v + (hi ? 8 : 0);
    utile[wave][m * USTRIDE + n] = U[v];
  }
  __syncthreads();

  const float cW = -P[3];
  const float G0 = Gv[0], G1 = Gv[1], G2 = Gv[2];
  const float G3 = Gv[3], G4 = Gv[4], G5 = Gv[5];
  // hardware rcp (<=1 ulp) instead of the ~50-instr IEEE div expansion
  const float i0 = __builtin_amdgcn_rcpf(Cv[0]);
  const float i1 = __builtin_amdgcn_rcpf(Cv[1]);
  const float i2 = __builtin_amdgcn_rcpf(Cv[2]);
  const float i3 = __builtin_amdgcn_rcpf(Cv[3]);
  const float i4 = __builtin_amdgcn_rcpf(Cv[4]);

  // 64 float4 groups per tile; lane owns groups {lane, lane+32}.
#pragma unroll
  for (int gi = 0; gi < 2; ++gi) {
    const int g   = lane + gi * 32;
    const int row = g >> 2;          // 0..15
    const int cg  = g & 3;           // 0..3
    const size_t idx = (size_t)(J0 + row) * TW2 + (size_t)(K0 + cg * 4);

    const v4f uv = *(const v4f*)&utile[wave][row * USTRIDE + cg * 4];
    const v4f w  = *(const v4f*)(Wm + idx);
    const v4f bi = *(const v4f*)(bias + idx);
    const v4f c0 = *(const v4f*)(chem + idx);
    const v4f c1 = *(const v4f*)(chem + NELEM + idx);
    const v4f c2 = *(const v4f*)(chem + 2 * NELEM + idx);
    const v4f c3 = *(const v4f*)(chem + 3 * NELEM + idx);
    const v4f c4 = *(const v4f*)(chem + 4 * NELEM + idx);

    v4f t;
    t.x = fast_tanh(__builtin_fmaf(cW, w.x, uv.x) + bi.x);
    t.y = fast_tanh(__builtin_fmaf(cW, w.y, uv.y) + bi.y);
    t.z = fast_tanh(__builtin_fmaf(cW, w.z, uv.z) + bi.z);
    t.w = fast_tanh(__builtin_fmaf(cW, w.w, uv.w) + bi.w);

    v4f o0, o1, o2, o3, o4;
    o0.x = c0.x + (t.x + G1 * (c1.x - c0.x)) * i0;
    o0.y = c0.y + (t.y + G1 * (c1.y - c0.y)) * i0;
    o0.z = c0.z + (t.z + G1 * (c1.z - c0.z)) * i0;
    o0.w = c0.w + (t.w + G1 * (c1.w - c0.w)) * i0;

    o1.x = c1.x + (G0 * (c0.x - c1.x) + G2 * (c2.x - c1.x)) * i1;
    o1.y = c1.y + (G0 * (c0.y - c1.y) + G2 * (c2.y - c1.y)) * i1;
    o1.z = c1.z + (G0 * (c0.z - c1.z) + G2 * (c2.z - c1.z)) * i1;
    o1.w = c1.w + (G0 * (c0.w - c1.w) + G2 * (c2.w - c1.w)) * i1;

    o2.x = c2.x + (G1 * (c1.x - c2.x) + G3 * (c3.x - c2.x)) * i2;
    o2.y = c2.y + (G1 * (c1.y - c2.y) + G3 * (c3.y - c2.y)) * i2;
    o2.z = c2.z + (G1 * (c1.z - c2.z) + G3 * (c3.z - c2.z)) * i2;
    o2.w = c2.w + (G1 * (c1.w - c2.w) + G3 * (c3.w - c2.w)) * i2;

    o3.x = c3.x + (G2 * (c2.x - c3.x) + G4 * (c4.x - c3.x)) * i3;
    o3.y = c3.y + (G2 * (c2.y - c3.y) + G4 * (c4.y - c3.y)) * i3;
    o3.z = c3.z + (G2 * (c2.z - c3.z) + G4 * (c4.z - c3.z)) * i3;
    o3.w = c3.w + (G2 * (c2.w - c3.w) + G4 * (c4.w - c3.w)) * i3;

    o4.x = c4.x + (G3 * (c3.x - c4.x) - G5 * c4.x) * i4;
    o4.y = c4.y + (G3 * (c3.y - c4.y) - G5 * c4.y) * i4;
    o4.z = c4.z + (G3 * (c3.z - c4.z) - G5 * c4.z) * i4;
    o4.w = c4.w + (G3 * (c3.w - c4.w) - G5 * c4.w) * i4;

    __builtin_nontemporal_store(o0, (v4f*)(out + idx));
    __builtin_nontemporal_store(o1, (v4f*)(out + NELEM + idx));
    __builtin_nontemporal_store(o2, (v4f*)(out + 2 * NELEM + idx));
    __builtin_nontemporal_store(o3, (v4f*)(out + 3 * NELEM + idx));
    __builtin_nontemporal_store(o4, (v4f*)(out + 4 * NELEM + idx));
  }
}

// ---------------------------------------------------------------------------
extern "C" void kernel_launch(void* const* d_in, const int* in_sizes, int n_in,
                              void* d_out, int out_size, void* d_ws, size_t ws_size,
                              hipStream_t stream) {
  const float* a0   = (const float*)d_in[0];
  const float* a1   = (const float*)d_in[1];
  const float* e0   = (const float*)d_in[2];
  const float* e1   = (const float*)d_in[3];
  const float* Wm   = (const float*)d_in[4];
  const float* chem = (const float*)d_in[5];
  const float* P    = (const float*)d_in[6];
  const float* bias = (const float*)d_in[7];
  const float* Cv   = (const float*)d_in[8];
  const float* Gv   = (const float*)d_in[9];
  float* out = (float*)d_out;

  float* ws   = (float*)d_ws;
  float* r    = ws;                 // 4096
  float* scal = ws + TW2;           // [0]=s67 [1]=s8 (+pad to 16)
  float* xcol = ws + TW2 + 16;      // 4096
  float* ycol = xcol + TW2;         // 4096
  float* zcol = ycol + TW2;         // 4096

  // zero the atomic accumulators (r + scalars) every call — deterministic
  (void)hipMemsetAsync(r, 0, (TW2 + 16) * sizeof(float), stream);

  reduce_pass<<<128, 256, 0, stream>>>(Wm, a1, e1, a0, e0, r, scal);
  prep_pass<<<1, 256, 0, stream>>>(a0, e0, a1, P, r, scal, xcol, ycol, zcol);

  const int tiles  = (TW1 / 16) * (TW2 / 16);  // 65536 tiles, 1 wave each
  const int blocks = tiles / 8;                // 8 waves per 256-thread block
  benna_main<<<blocks, 256, 0, stream>>>(Wm, bias, chem, a1, e1,
                                         xcol, ycol, zcol, P, Cv, Gv, out);
}